// RNNWithMarkovNet_54425825575652
// MI455X (gfx1250) — compile-verified
//
#include <hip/hip_runtime.h>
#include <hip/hip_bf16.h>

#define SDIM 64
#define EDIM 128
#define TSTEP 256
#define BATCH 32
#define VDIM 32000
#define NT (VDIM / 16)   // 2000 N-tiles

typedef float v2f __attribute__((ext_vector_type(2)));
typedef float v8f __attribute__((ext_vector_type(8)));
typedef int   v4i __attribute__((vector_size(4 * sizeof(int))));   // matches builtin param type

#define AS1 __attribute__((address_space(1)))
#define AS3 __attribute__((address_space(3)))

// ---- async global->LDS b128 copy (ASYNCcnt-tracked), builtin if present, else inline asm ----
__device__ __forceinline__ void async_copy_b128(const float* g, float* l) {
#if __has_builtin(__builtin_amdgcn_global_load_async_to_lds_b128)
    __builtin_amdgcn_global_load_async_to_lds_b128((AS1 v4i*)g, (AS3 v4i*)l, 0, 0);
#else
    unsigned lofs = (unsigned)(uintptr_t)(AS3 float*)l;
    unsigned long long ga = (unsigned long long)(uintptr_t)g;
    asm volatile("global_load_async_to_lds_b128 %0, %1, off" :: "v"(lofs), "v"(ga) : "memory");
#endif
}

__device__ __forceinline__ void wait_async0() {
#if __has_builtin(__builtin_amdgcn_s_wait_asynccnt)
    __builtin_amdgcn_s_wait_asynccnt(0);
#else
    asm volatile("s_wait_asynccnt 0x0" ::: "memory");
#endif
}

__device__ __forceinline__ const float* slab_ptr(const float* latent, const int* zi, int f, int b) {
    int idx = f * BATCH + b;
    return latent + (size_t)zi[idx >> 2] * 32768 + (size_t)(idx & 3) * 8192;
}

// ---------------- Kernel 1: scores GEMM + column softmax + att0 ----------------
__global__ __launch_bounds__(256) void prep_kernel(const float* __restrict__ latent,
                                                   const int* __restrict__ zi,
                                                   float* __restrict__ xtran,
                                                   float* __restrict__ att0) {
    __shared__ float kbuf[SDIM * EDIM];   // 32 KB
    __shared__ float sc[SDIM * SDIM];     // 16 KB
    __shared__ float red[SDIM];
    __shared__ float m_sh, s_sh;

    const int b   = blockIdx.x;
    const int tid = threadIdx.x;

    const float* q  = slab_ptr(latent, zi, 0, b);
    const float* k  = slab_ptr(latent, zi, 1, b);
    const float* xi = slab_ptr(latent, zi, 3, b);

    for (int i = tid; i < SDIM * EDIM; i += 256) kbuf[i] = k[i];
    __syncthreads();

    // scores[s][t] = (1/sqrt(128)) * sum_e q[s][e]*k[t][e]
    const int s  = tid >> 2;
    const int t0 = tid & 3;
    float acc[16];
#pragma unroll
    for (int i = 0; i < 16; i++) acc[i] = 0.f;
    for (int e = 0; e < EDIM; e++) {
        float qv = q[s * EDIM + e];
#pragma unroll
        for (int i = 0; i < 16; i++) acc[i] += qv * kbuf[(t0 + 4 * i) * EDIM + e];
    }
    const float scale = 0.08838834764831845f;  // 1/sqrt(128)
#pragma unroll
    for (int i = 0; i < 16; i++) sc[s * SDIM + (t0 + 4 * i)] = acc[i] * scale;

    if (tid < SDIM) red[tid] = xi[tid * EDIM];  // xi[:, s, 0]
    __syncthreads();

    // softmax over s (axis=1) for each column t
    if (tid < SDIM) {
        const int t = tid;
        float m = -INFINITY;
        for (int ss = 0; ss < SDIM; ss++) m = fmaxf(m, sc[ss * SDIM + t]);
        float sum = 0.f;
        for (int ss = 0; ss < SDIM; ss++) sum += __expf(sc[ss * SDIM + t] - m);
        float inv = 1.f / sum;
        for (int ss = 0; ss < SDIM; ss++)
            xtran[((size_t)b * SDIM + ss) * SDIM + t] = __expf(sc[ss * SDIM + t] - m) * inv;
    }
    if (tid == 0) {
        float m = -INFINITY;
        for (int ss = 0; ss < SDIM; ss++) m = fmaxf(m, red[ss]);
        float sum = 0.f;
        for (int ss = 0; ss < SDIM; ss++) sum += __expf(red[ss] - m);
        m_sh = m; s_sh = sum;
    }
    __syncthreads();
    if (tid < SDIM) att0[b * SDIM + tid] = __expf(red[tid] - m_sh) / s_sh;
}

// ---------------- Kernel 2: 256-step Markov scan, emits zs ----------------
__global__ __launch_bounds__(256) void scan_kernel(const float* __restrict__ latent,
                                                   const int* __restrict__ zi,
                                                   const float* __restrict__ xtran,
                                                   const float* __restrict__ att0,
                                                   float* __restrict__ zs) {
    __shared__ float xv[SDIM * EDIM];   // 32 KB
    __shared__ float tr[SDIM * SDIM];   // 16 KB
    __shared__ float att[SDIM];
    __shared__ float attn[SDIM];

    const int b   = blockIdx.x;
    const int tid = threadIdx.x;

    const float* v = slab_ptr(latent, zi, 2, b);
    for (int i = tid; i < SDIM * EDIM; i += 256) xv[i] = v[i];
    for (int i = tid; i < SDIM * SDIM; i += 256) tr[i] = xtran[(size_t)b * SDIM * SDIM + i];
    if (tid < SDIM) att[tid] = att0[b * SDIM + tid];
    __syncthreads();

    for (int t = 0; t < TSTEP; t++) {
        if (tid < EDIM) {
            float a = 0.f;
            for (int ss = 0; ss < SDIM; ss++) a += att[ss] * xv[ss * EDIM + tid];
            zs[((size_t)(b * TSTEP + t)) * EDIM + tid] = a;
        } else if (tid < EDIM + SDIM) {
            const int tp = tid - EDIM;
            float a = 0.f;
            for (int ss = 0; ss < SDIM; ss++) a += att[ss] * tr[ss * SDIM + tp];
            attn[tp] = a;
        }
        __syncthreads();
        if (tid < SDIM) att[tid] = attn[tid];
        __syncthreads();
    }
}

// ---------------- Kernel 3: fused zs @ vocab_w^T + online logsumexp ----------------
// 64 WGs x 8 waves; wave owns a 16-row strip of zs (A resident in 64 VGPRs).
// Vocab tiles staged once per WG into double-buffered LDS via async global->LDS
// (ASYNCcnt), then each wave runs 32 x v_wmma_f32_16x16x4_f32 off ds_load_b64 fragments.
__global__ __launch_bounds__(256) void vocab_lse_kernel(const float* __restrict__ zs,
                                                        const float* __restrict__ w,
                                                        const float* __restrict__ bias,
                                                        float* __restrict__ lse) {
    __shared__ float ldsB[2][16 * EDIM];   // 2 x 8 KB tile buffers

    const int tid  = threadIdx.x;
    const int lane = tid & 31;
    const int wave = tid >> 5;
    const int m0   = blockIdx.x * 128 + wave * 16;
    const int rowA = m0 + (lane & 15);
    const int kofs = (lane >> 4) << 1;   // lanes 16-31 hold K = k0+2, k0+3

    // A fragments: 16x128 strip, 32 chunks of 16x4 (v2f per lane each)
    v2f a[32];
#pragma unroll
    for (int kk = 0; kk < 32; kk++)
        a[kk] = *(const v2f*)(zs + (size_t)rowA * EDIM + kk * 4 + kofs);

    float m8[8], s8[8];
#pragma unroll
    for (int j = 0; j < 8; j++) { m8[j] = -1e30f; s8[j] = 0.f; }

    // stage tile 0 (each of 256 threads copies 2 x b128 = 32B, fully coalesced)
    {
        const float* src = w + (size_t)tid * 8;
        async_copy_b128(src,     &ldsB[0][tid * 8]);
        async_copy_b128(src + 4, &ldsB[0][tid * 8 + 4]);
    }
    wait_async0();
    __syncthreads();

    for (int nt = 0; nt < NT; nt++) {
        const int buf = nt & 1;
        if (nt + 1 < NT) {  // prefetch next tile into the other buffer
            const float* src = w + (size_t)(nt + 1) * 16 * EDIM + tid * 8;
            async_copy_b128(src,     &ldsB[buf ^ 1][tid * 8]);
            async_copy_b128(src + 4, &ldsB[buf ^ 1][tid * 8 + 4]);
        }

        const float* brow = &ldsB[buf][(lane & 15) * EDIM + kofs];
        v8f c = {0.f, 0.f, 0.f, 0.f, 0.f, 0.f, 0.f, 0.f};
#pragma unroll
        for (int kk = 0; kk < 32; kk++) {
            v2f bf = *(const v2f*)(brow + kk * 4);   // ds_load_b64 fragment
            c = __builtin_amdgcn_wmma_f32_16x16x4_f32(false, a[kk], false, bf,
                                                      (short)0, c, false, false);
        }

        const float bv = bias[nt * 16 + (lane & 15)];
#pragma unroll
        for (int j = 0; j < 8; j++) {
            float logit = c[j] + bv;   // lane holds row m0 + j + 8*(lane>>4), col nt*16 + (lane&15)
            if (logit <= m8[j]) {
                s8[j] += __expf(logit - m8[j]);
            } else {
                s8[j] = s8[j] * __expf(m8[j] - logit) + 1.f;
                m8[j] = logit;
            }
        }

        wait_async0();      // own prefetch complete
        __syncthreads();    // everyone done reading buf + everyone's prefetch visible
    }

    // butterfly reduce (max, sumexp) across the 16 lanes holding each row
#pragma unroll
    for (int off = 1; off < 16; off <<= 1) {
#pragma unroll
        for (int j = 0; j < 8; j++) {
            float mo = __shfl_xor(m8[j], off, 32);
            float so = __shfl_xor(s8[j], off, 32);
            float mn = fmaxf(m8[j], mo);
            s8[j] = s8[j] * __expf(m8[j] - mn) + so * __expf(mo - mn);
            m8[j] = mn;
        }
    }
    if ((lane & 15) == 0) {
        const int half = lane >> 4;
#pragma unroll
        for (int j = 0; j < 8; j++)
            lse[m0 + half * 8 + j] = m8[j] + __logf(s8[j]);
    }
}

// ---------------- Kernel 4: target logit + final yp ----------------
__global__ __launch_bounds__(32) void target_kernel(const float* __restrict__ zs,
                                                    const float* __restrict__ w,
                                                    const float* __restrict__ bias,
                                                    const int* __restrict__ y,
                                                    const float* __restrict__ lse,
                                                    float* __restrict__ yp) {
    const int r    = blockIdx.x;      // r = b*256 + t
    const int lane = threadIdx.x;
    const int v    = y[r];
    float acc = 0.f;
    for (int e = lane; e < EDIM; e += 32)
        acc += zs[(size_t)r * EDIM + e] * w[(size_t)v * EDIM + e];
#pragma unroll
    for (int off = 16; off > 0; off >>= 1) acc += __shfl_down(acc, off, 32);
    if (lane == 0) yp[r] = acc + bias[v] - lse[r];
}

extern "C" void kernel_launch(void* const* d_in, const int* in_sizes, int n_in,
                              void* d_out, int out_size, void* d_ws, size_t ws_size,
                              hipStream_t stream) {
    const float* latent  = (const float*)d_in[0];
    const float* vocab_w = (const float*)d_in[1];
    const float* vocab_b = (const float*)d_in[2];
    const int*   zi      = (const int*)d_in[3];
    const int*   y       = (const int*)d_in[4];
    float*       yp      = (float*)d_out;

    float* ws    = (float*)d_ws;
    float* xtran = ws;                                  // 32*64*64   = 131072 floats
    float* att0  = ws + 131072;                         // 32*64      = 2048
    float* zs    = ws + 133120;                         // 8192*128   = 1048576
    float* lse   = ws + 1181696;                        // 8192

    prep_kernel<<<BATCH, 256, 0, stream>>>(latent, zi, xtran, att0);
    scan_kernel<<<BATCH, 256, 0, stream>>>(latent, zi, xtran, att0, zs);
    vocab_lse_kernel<<<64, 256, 0, stream>>>(zs, vocab_w, vocab_b, lse);
    target_kernel<<<8192, 32, 0, stream>>>(zs, vocab_w, vocab_b, y, lse, yp);
}